// TemporalGNN_65661460021945
// MI455X (gfx1250) — compile-verified
//
#include <hip/hip_runtime.h>
#include <cstddef>

// ---------------------------------------------------------------------------
// Types for CDNA5 WMMA f32 16x16x4
// ---------------------------------------------------------------------------
typedef __attribute__((ext_vector_type(2))) float v2f;
typedef __attribute__((ext_vector_type(8))) float v8f;

// Order-preserving float<->uint encoding so f32 max can use u32 atomicMax.
__device__ __forceinline__ unsigned fenc(float f) {
  unsigned u = __float_as_uint(f);
  return (u & 0x80000000u) ? ~u : (u | 0x80000000u);
}
__device__ __forceinline__ float fdec(unsigned u) {
  return __uint_as_float((u & 0x80000000u) ? (u ^ 0x80000000u) : ~u);
}

__device__ __forceinline__ float dot4(float4 a, float4 b) {
  return a.x * b.x + a.y * b.y + a.z * b.z + a.w * b.w;
}

// ---------------------------------------------------------------------------
// Zero-fill (graph-capture-safe init; harness does not re-poison between runs)
// ---------------------------------------------------------------------------
__global__ void zero_buf(float4* __restrict__ p, size_t n4) {
  size_t i = (size_t)blockIdx.x * blockDim.x + threadIdx.x;
  if (i < n4) p[i] = make_float4(0.f, 0.f, 0.f, 0.f);
}

// ---------------------------------------------------------------------------
// x_i[e][f] = emb[x[row[e]]][f] + t[e]*w_t[f] + b_t[f]     (16 threads/edge, f4)
// ---------------------------------------------------------------------------
__global__ void build_xi(const int* __restrict__ x, const int* __restrict__ row,
                         const float* __restrict__ t, const float* __restrict__ emb,
                         const float* __restrict__ w_t, const float* __restrict__ b_t,
                         float* __restrict__ xi, int E) {
  int gid = blockIdx.x * blockDim.x + threadIdx.x;
  int e = gid >> 4;
  int c = (gid & 15) * 4;
  if (e >= E) return;
  int node = x[row[e]];
  float te = t[e];
  float4 em = *(const float4*)(emb + (size_t)node * 64 + c);
  float4 wv = *(const float4*)(w_t + c);
  float4 bv = *(const float4*)(b_t + c);
  float4 r;
  r.x = em.x + te * wv.x + bv.x;
  r.y = em.y + te * wv.y + bv.y;
  r.z = em.z + te * wv.z + bv.z;
  r.w = em.w + te * wv.w + bv.w;
  *(float4*)(xi + (size_t)e * 64 + c) = r;
}

// ---------------------------------------------------------------------------
// WMMA f32 GEMM: C[M,N] = act(A[M,K] @ B[K,N] (+bias)).
// Block = NWAVES waves; one 16-row M tile per block, A staged in LDS,
// each wave covers N/16/NWAVES 16x16 tiles via v_wmma_f32_16x16x4_f32.
// GATHER path builds A rows as concat(Asrc[rowi[e]], Asrc[coli[e]]) (K=128).
// Full tiles (the common case: M % 16 == 0) take a branch-free store path.
// ---------------------------------------------------------------------------
template <int K, int N, int NWAVES, bool GATHER, bool RELU, bool BIAS>
__global__ __launch_bounds__(NWAVES * 32) void gemm_wmma(
    const float* __restrict__ A, const float* __restrict__ B,
    const float* __restrict__ bias, float* __restrict__ C,
    const int* __restrict__ rowi, const int* __restrict__ coli, int M) {
  __shared__ float Alds[16][K + 2];  // +2 pad: conflict-free ds_load_b64 fragments
  const int mBase = blockIdx.x * 16;
  const int tid = threadIdx.x;

  if constexpr (GATHER) {
    for (int idx = tid * 4; idx < 16 * K; idx += NWAVES * 32 * 4) {
      int r = idx / K, c = idx - r * K;
      int e = mBase + r;
      if (e >= M) e = M - 1;
      int node = (c < K / 2) ? rowi[e] : coli[e];
      int cc = c & (K / 2 - 1);
      float4 v = *(const float4*)(A + (size_t)node * (K / 2) + cc);
      Alds[r][c] = v.x; Alds[r][c + 1] = v.y; Alds[r][c + 2] = v.z; Alds[r][c + 3] = v.w;
    }
  } else {
    for (int idx = tid * 4; idx < 16 * K; idx += NWAVES * 32 * 4) {
      int r = idx / K, c = idx - r * K;
      int gr = mBase + r;
      if (gr >= M) gr = M - 1;
      float4 v = *(const float4*)(A + (size_t)gr * K + c);
      Alds[r][c] = v.x; Alds[r][c + 1] = v.y; Alds[r][c + 2] = v.z; Alds[r][c + 3] = v.w;
    }
  }
  __syncthreads();

  const int lane = tid & 31;
  const int wave = tid >> 5;
  const int mrow = lane & 15;   // A: M index for this lane
  const int kb = lane >> 4;     // A/B: K-pair select (K+0,1 vs K+2,3)
  const bool fullTile = (mBase + 16 <= M);  // uniform

  for (int nt = wave; nt < N / 16; nt += NWAVES) {
    const int n = nt * 16 + (lane & 15);
    v8f acc = {0.f, 0.f, 0.f, 0.f, 0.f, 0.f, 0.f, 0.f};
#pragma unroll
    for (int k0 = 0; k0 < K; k0 += 4) {
      v2f a;
      a.x = Alds[mrow][k0 + 2 * kb];
      a.y = Alds[mrow][k0 + 2 * kb + 1];
      v2f b;
      b.x = B[(size_t)(k0 + 2 * kb) * N + n];
      b.y = B[(size_t)(k0 + 2 * kb + 1) * N + n];
      acc = __builtin_amdgcn_wmma_f32_16x16x4_f32(false, a, false, b, (short)0,
                                                  acc, false, false);
    }
    const float bn = BIAS ? bias[n] : 0.f;   // one load per column
    float* cp = C + (size_t)(mBase + 8 * (lane >> 4)) * N + n;
    if (fullTile) {
#pragma unroll
      for (int r = 0; r < 8; r++) {
        float v = acc[r] + bn;
        if (RELU) v = fmaxf(v, 0.f);
        cp[(size_t)r * N] = v;               // branch-free, coalesced
      }
    } else {
#pragma unroll
      for (int r = 0; r < 8; r++) {
        int m = r + 8 * (lane >> 4);
        if (mBase + m < M) {
          float v = acc[r] + bn;
          if (RELU) v = fmaxf(v, 0.f);
          C[(size_t)(mBase + m) * N + n] = v;
        }
      }
    }
  }
}

// ---------------------------------------------------------------------------
// Attention logits, H=4: al_s[e][h] = sum_d h1[e][h*64+d]*a_src[h][d]
// one wave per edge row of 256 floats; 8-lane group reduction per head.
// ---------------------------------------------------------------------------
__global__ void att_logits_h4(const float* __restrict__ h1,
                              const float* __restrict__ a_s,
                              const float* __restrict__ a_d,
                              float* __restrict__ al_s, float* __restrict__ al_d,
                              int E) {
  int wid = (int)(((size_t)blockIdx.x * blockDim.x + threadIdx.x) >> 5);
  int lane = threadIdx.x & 31;
  if (wid >= E) return;
  const float4* hp = (const float4*)(h1 + (size_t)wid * 256);
  float4 h0 = hp[lane * 2], h1v = hp[lane * 2 + 1];
  const float4* sp = (const float4*)a_s;
  const float4* dp = (const float4*)a_d;
  float ps = dot4(h0, sp[lane * 2]) + dot4(h1v, sp[lane * 2 + 1]);
  float pd = dot4(h0, dp[lane * 2]) + dot4(h1v, dp[lane * 2 + 1]);
#pragma unroll
  for (int m = 1; m <= 4; m <<= 1) {
    ps += __shfl_xor(ps, m);
    pd += __shfl_xor(pd, m);
  }
  if ((lane & 7) == 0) {
    al_s[(size_t)wid * 4 + (lane >> 3)] = ps;
    al_d[(size_t)wid * 4 + (lane >> 3)] = pd;
  }
}

// H=1 over 64-float rows: full-wave reduction.
__global__ void att_logits_h1(const float* __restrict__ h2,
                              const float* __restrict__ a_s,
                              const float* __restrict__ a_d,
                              float* __restrict__ al_s, float* __restrict__ al_d,
                              int E) {
  int wid = (int)(((size_t)blockIdx.x * blockDim.x + threadIdx.x) >> 5);
  int lane = threadIdx.x & 31;
  if (wid >= E) return;
  float2 h = *(const float2*)(h2 + (size_t)wid * 64 + lane * 2);
  float2 s = *(const float2*)(a_s + lane * 2);
  float2 d = *(const float2*)(a_d + lane * 2);
  float ps = h.x * s.x + h.y * s.y;
  float pd = h.x * d.x + h.y * d.y;
#pragma unroll
  for (int m = 1; m <= 16; m <<= 1) {
    ps += __shfl_xor(ps, m);
    pd += __shfl_xor(pd, m);
  }
  if (lane == 0) {
    al_s[wid] = ps;
    al_d[wid] = pd;
  }
}

// ---------------------------------------------------------------------------
// e = leaky_relu(al_s[s]+al_d[d], 0.2); segment-max via encoded u32 atomicMax.
// Entries: [0,E) = edges, [E,2E) = self loops.
// ---------------------------------------------------------------------------
__global__ void edge_logits(const int* __restrict__ row, const int* __restrict__ col,
                            const float* __restrict__ al_s, const float* __restrict__ al_d,
                            float* __restrict__ ebuf, unsigned* __restrict__ mbuf,
                            int E, int H) {
  int i = blockIdx.x * blockDim.x + threadIdx.x;
  if (i >= 2 * E) return;
  int s = (i < E) ? row[i] : (i - E);
  int d = (i < E) ? col[i] : (i - E);
  for (int h = 0; h < H; h++) {
    float v = al_s[(size_t)s * H + h] + al_d[(size_t)d * H + h];
    v = (v >= 0.f) ? v : 0.2f * v;
    ebuf[(size_t)i * H + h] = v;
    atomicMax(&mbuf[(size_t)d * H + h], fenc(v));
  }
}

// ex = exp(e - m[d]); segment-sum via f32 atomicAdd; ebuf overwritten with ex.
__global__ void exp_sum(const int* __restrict__ col, float* __restrict__ ebuf,
                        const unsigned* __restrict__ mbuf, float* __restrict__ sbuf,
                        int E, int H) {
  int i = blockIdx.x * blockDim.x + threadIdx.x;
  if (i >= 2 * E) return;
  int d = (i < E) ? col[i] : (i - E);
  for (int h = 0; h < H; h++) {
    float ex = expf(ebuf[(size_t)i * H + h] - fdec(mbuf[(size_t)d * H + h]));
    ebuf[(size_t)i * H + h] = ex;
    atomicAdd(&sbuf[(size_t)d * H + h], ex);
  }
}

// Aggregation H=4: out[d] += h1[s] * alpha  (wave per entry, 8 floats/lane)
__global__ void agg_h4(const int* __restrict__ row, const int* __restrict__ col,
                       const float* __restrict__ h1, const float* __restrict__ ebuf,
                       const float* __restrict__ sbuf, float* __restrict__ out, int E) {
  int wid = (int)(((size_t)blockIdx.x * blockDim.x + threadIdx.x) >> 5);
  int lane = threadIdx.x & 31;
  if (wid >= 2 * E) return;
  int s = (wid < E) ? row[wid] : (wid - E);
  int d = (wid < E) ? col[wid] : (wid - E);
  int head = lane >> 3;
  float alpha = ebuf[(size_t)wid * 4 + head] / (sbuf[(size_t)d * 4 + head] + 1e-16f);
  const float4* hp = (const float4*)(h1 + (size_t)s * 256);
  float4 a = hp[lane * 2], b = hp[lane * 2 + 1];
  float* op = out + (size_t)d * 256 + lane * 8;
  atomicAdd(op + 0, a.x * alpha);
  atomicAdd(op + 1, a.y * alpha);
  atomicAdd(op + 2, a.z * alpha);
  atomicAdd(op + 3, a.w * alpha);
  atomicAdd(op + 4, b.x * alpha);
  atomicAdd(op + 5, b.y * alpha);
  atomicAdd(op + 6, b.z * alpha);
  atomicAdd(op + 7, b.w * alpha);
}

// Aggregation H=1: 64 floats, 2/lane.
__global__ void agg_h1(const int* __restrict__ row, const int* __restrict__ col,
                       const float* __restrict__ h2, const float* __restrict__ ebuf,
                       const float* __restrict__ sbuf, float* __restrict__ out, int E) {
  int wid = (int)(((size_t)blockIdx.x * blockDim.x + threadIdx.x) >> 5);
  int lane = threadIdx.x & 31;
  if (wid >= 2 * E) return;
  int s = (wid < E) ? row[wid] : (wid - E);
  int d = (wid < E) ? col[wid] : (wid - E);
  float alpha = ebuf[wid] / (sbuf[d] + 1e-16f);
  float2 hv = *(const float2*)(h2 + (size_t)s * 64 + lane * 2);
  float* op = out + (size_t)d * 64 + lane * 2;
  atomicAdd(op + 0, hv.x * alpha);
  atomicAdd(op + 1, hv.y * alpha);
}

// buf[i] = act(buf[i] + bias[i % N]), N power-of-two, float4.
__global__ void bias_act(float* __restrict__ buf, const float* __restrict__ bias,
                         size_t total, int N, int relu) {
  size_t i4 = ((size_t)blockIdx.x * blockDim.x + threadIdx.x) * 4;
  if (i4 >= total) return;
  float4 v = *(float4*)(buf + i4);
  int c = (int)(i4 & (size_t)(N - 1));
  v.x += bias[c]; v.y += bias[c + 1]; v.z += bias[c + 2]; v.w += bias[c + 3];
  if (relu) {
    v.x = fmaxf(v.x, 0.f); v.y = fmaxf(v.y, 0.f);
    v.z = fmaxf(v.z, 0.f); v.w = fmaxf(v.w, 0.f);
  }
  *(float4*)(buf + i4) = v;
}

// out[e] = sigmoid(dot(z[e,:64], w) + b)
__global__ void final_dot(const float* __restrict__ z, const float* __restrict__ w,
                          const float* __restrict__ b, float* __restrict__ out, int E) {
  int wid = (int)(((size_t)blockIdx.x * blockDim.x + threadIdx.x) >> 5);
  int lane = threadIdx.x & 31;
  if (wid >= E) return;
  float2 zv = *(const float2*)(z + (size_t)wid * 64 + lane * 2);
  float2 wv = *(const float2*)(w + lane * 2);
  float p = zv.x * wv.x + zv.y * wv.y;
#pragma unroll
  for (int m = 1; m <= 16; m <<= 1) p += __shfl_xor(p, m);
  if (lane == 0) out[wid] = 1.f / (1.f + expf(-(p + b[0])));
}

// ---------------------------------------------------------------------------
extern "C" void kernel_launch(void* const* d_in, const int* in_sizes, int n_in,
                              void* d_out, int out_size, void* d_ws, size_t ws_size,
                              hipStream_t stream) {
  const int E = in_sizes[1] / 2;  // 800000
  const int* x = (const int*)d_in[0];
  const int* row = (const int*)d_in[1];
  const int* col = row + E;
  const float* tfeat = (const float*)d_in[2];
  const float* emb = (const float*)d_in[3];
  const float* w_t = (const float*)d_in[4];
  const float* b_t = (const float*)d_in[5];
  const float* w_g1 = (const float*)d_in[6];
  const float* a1_src = (const float*)d_in[7];
  const float* a1_dst = (const float*)d_in[8];
  const float* b_g1 = (const float*)d_in[9];
  const float* w_g2 = (const float*)d_in[10];
  const float* a2_src = (const float*)d_in[11];
  const float* a2_dst = (const float*)d_in[12];
  const float* b_g2 = (const float*)d_in[13];
  const float* w_p1 = (const float*)d_in[14];
  const float* b_p1 = (const float*)d_in[15];
  const float* w_p2 = (const float*)d_in[16];
  const float* b_p2 = (const float*)d_in[17];
  float* out = (float*)d_out;

  // ---- workspace layout (floats), regions time-multiplexed -----------------
  const size_t Ez = (size_t)E;
  float* W = (float*)d_ws;
  float* h1 = W;                 // [E,256] layer-1 GEMM out; reused later:
  float* h2lin = W;              //   [E,64]  layer-2 GEMM out (h1 dead)
  float* out2 = W + 64 * Ez;     //   [E,64]  layer-2 aggregation -> h2
  float* zbuf = W + 128 * Ez;    //   [E,64]  MLP hidden
  float* out1 = W + 256 * Ez;    // [E,256] layer-1 aggregation / GEMM2 input
  float* small = W + 512 * Ez;   // 64E floats, carved below
  float* xi = small;             // [E,64], dead after GEMM1 (overlaps smalls OK)
  float* al1_s = small;          // 4E  (written after xi dead)
  float* al1_d = small + 4 * Ez; // 4E
  float* e1 = small + 8 * Ez;    // 8E (2E entries x H=4)
  unsigned* m1 = (unsigned*)(small + 16 * Ez);  // 4E
  float* s1 = small + 20 * Ez;   // 4E
  float* al2_s = small + 24 * Ez;
  float* al2_d = small + 25 * Ez;
  float* e2 = small + 26 * Ez;   // 2E
  unsigned* m2 = (unsigned*)(small + 28 * Ez);
  float* s2 = small + 29 * Ez;

  const int TB = 256;
  const int nEnt = 2 * E;  // edges + self loops
  const int blkEnt = (nEnt + TB - 1) / TB;         // thread per entry
  const int blkWaveE = (E * 32 + TB - 1) / TB;     // wave per edge
  const int blkWaveEnt = (nEnt * 32 + TB - 1) / TB;
  const int M16 = (E + 15) / 16;

  // zero out1 accumulator (independent region)
  {
    size_t n4 = (256 * Ez) / 4;
    zero_buf<<<(unsigned)((n4 + TB - 1) / TB), TB, 0, stream>>>((float4*)out1, n4);
  }

  // 1) x_i
  build_xi<<<(E * 16 + TB - 1) / TB, TB, 0, stream>>>(x, row, tfeat, emb, w_t, b_t, xi, E);

  // 2) GEMM1: h1 = x_i @ w_g1   [E,64]x[64,256]
  gemm_wmma<64, 256, 8, false, false, false><<<M16, 8 * 32, 0, stream>>>(
      xi, w_g1, nullptr, h1, nullptr, nullptr, E);

  // xi now dead -> safe to zero m1/s1 inside its region
  {
    size_t n4 = (8 * Ez) / 4;  // m1 (4E u32) + s1 (4E f32), contiguous
    zero_buf<<<(unsigned)((n4 + TB - 1) / TB), TB, 0, stream>>>((float4*)m1, n4);
  }

  // 3) layer-1 attention logits
  att_logits_h4<<<blkWaveE, TB, 0, stream>>>(h1, a1_src, a1_dst, al1_s, al1_d, E);
  // 4) edge logits + segment max
  edge_logits<<<blkEnt, TB, 0, stream>>>(row, col, al1_s, al1_d, e1, m1, E, 4);
  // 5) exp + segment sum
  exp_sum<<<blkEnt, TB, 0, stream>>>(col, e1, m1, s1, E, 4);
  // 6) aggregate into out1
  agg_h4<<<blkWaveEnt, TB, 0, stream>>>(row, col, h1, e1, s1, out1, E);
  // 7) out1 = relu(out1 + b_g1)
  bias_act<<<(unsigned)((64 * Ez + TB - 1) / TB), TB, 0, stream>>>(out1, b_g1, 256 * Ez, 256, 1);

  // 8) GEMM2: h2lin = out1 @ w_g2   [E,256]x[256,64]   (h1 region dead)
  gemm_wmma<256, 64, 4, false, false, false><<<M16, 4 * 32, 0, stream>>>(
      out1, w_g2, nullptr, h2lin, nullptr, nullptr, E);

  // zero layer-2 accumulator + m2/s2
  {
    size_t n4 = (64 * Ez) / 4;
    zero_buf<<<(unsigned)((n4 + TB - 1) / TB), TB, 0, stream>>>((float4*)out2, n4);
    size_t n4b = (2 * Ez) / 4;  // m2 + s2 contiguous
    zero_buf<<<(unsigned)((n4b + TB - 1) / TB), TB, 0, stream>>>((float4*)m2, n4b);
  }

  // 9) layer-2 attention (H=1)
  att_logits_h1<<<blkWaveE, TB, 0, stream>>>(h2lin, a2_src, a2_dst, al2_s, al2_d, E);
  edge_logits<<<blkEnt, TB, 0, stream>>>(row, col, al2_s, al2_d, e2, m2, E, 1);
  exp_sum<<<blkEnt, TB, 0, stream>>>(col, e2, m2, s2, E, 1);
  agg_h1<<<blkWaveEnt, TB, 0, stream>>>(row, col, h2lin, e2, s2, out2, E);
  // 10) h2 = out2 + b_g2 (concat=False over 1 head == identity)
  bias_act<<<(unsigned)((16 * Ez + TB - 1) / TB), TB, 0, stream>>>(out2, b_g2, 64 * Ez, 64, 0);

  // 11) GEMM3 (gather): z = relu(concat(h2[row],h2[col]) @ w_p1 + b_p1)
  gemm_wmma<128, 64, 4, true, true, true><<<M16, 4 * 32, 0, stream>>>(
      out2, w_p1, b_p1, zbuf, row, col, E);

  // 12) out = sigmoid(z @ w_p2 + b_p2)
  final_dot<<<blkWaveE, TB, 0, stream>>>(zbuf, w_p2, b_p2, out, E);

  (void)n_in; (void)out_size; (void)ws_size;
}